// kernel_generated_2_21887153341197
// MI455X (gfx1250) — compile-verified
//
#include <hip/hip_runtime.h>

typedef __attribute__((ext_vector_type(2))) float v2f;
typedef __attribute__((ext_vector_type(8))) float v8f;

#define KRED 144              // reduction length = 48 channels * 3 taps
#define PTOT 3136             // 56*56 spatial positions
#define PADW 146              // padded LDS row stride (floats), keeps b64 8B-aligned

// D[16x16] per wave, D = A(16x4) x B(4x16) + C chained over 36 K-steps.
// Reduction axis permuted to r = k*48 + j so each K-step has a uniform,
// compile-time tap index k: B addressing reduces to one base pointer per k
// plus compile-time immediate offsets, and halo masking to one cndmask.
__global__ __launch_bounds__(128) void conv_unfold_wmma_f32(
    const float* __restrict__ x,   // (48, 56, 56)
    const float* __restrict__ Wt,  // (192, 48, 3)
    float* __restrict__ out)       // (192, 56, 56)
{
    __shared__ float Ws[16 * PADW];

    const int tid = threadIdx.x;
    const int bx  = blockIdx.x;
    const int it  = bx % 12;          // output-channel tile
    const int pg  = bx / 12;          // group of 4 p-tiles (0..48)
    const int i0  = it * 16;

    // Stage 16x144 W slice into LDS with the reduction permute:
    // dest column r = k*48 + j  <-  source column j*3 + k
    for (int idx = tid; idx < 16 * KRED; idx += 128) {
        int row = idx / KRED;
        int col = idx - row * KRED;
        int j   = col / 3;
        int k   = col - 3 * j;
        Ws[row * PADW + k * 48 + j] = Wt[(i0 + row) * KRED + col];
    }
    __syncthreads();

    const int wave = tid >> 5;
    const int lane = tid & 31;
    const int half = lane >> 4;       // K split across half-waves
    const int Mrow = lane & 15;       // A row
    const int Ncol = lane & 15;       // B / D column

    const int p0  = (pg * 4 + wave) * 16;
    const int p   = p0 + Ncol;
    const int n   = p / 392;          // 392 = 7*56
    const int rem = p - n * 392;
    const int o   = rem / 56;         // index in the size-7 sub-axis
    const int m   = rem - o * 56;
    const int pbase = n * 392 + m;

    // Hoisted per-tap lane state: clamped base pointer + validity mask.
    const float* ptrk[3];
    bool validk[3];
    #pragma unroll
    for (int k = 0; k < 3; ++k) {
        int oo    = o + k - 1;
        validk[k] = (unsigned)oo < 7u;
        int ooc   = oo < 0 ? 0 : (oo > 6 ? 6 : oo);   // safe (never OOB) address
        ptrk[k]   = x + pbase + ooc * 56 + 2 * half * PTOT;
    }

    // Per-lane LDS base for the A fragment (b64: r = 4s+2*half, +1)
    const float* Wrow = &Ws[Mrow * PADW + 2 * half];

    v8f acc = {};
    #pragma unroll
    for (int s = 0; s < 36; ++s) {
        const int k  = s / 12;            // compile-time after full unroll
        const int jb = 4 * s - 48 * k;    // compile-time channel base

        v2f A;
        A.x = Wrow[4 * s];
        A.y = Wrow[4 * s + 1];

        float t0 = ptrk[k][(size_t)jb * PTOT];          // immediate offset
        float t1 = ptrk[k][(size_t)(jb + 1) * PTOT];    // immediate offset
        v2f B;
        B.x = validk[k] ? t0 : 0.0f;                    // halo zero: 1 cndmask
        B.y = validk[k] ? t1 : 0.0f;

        acc = __builtin_amdgcn_wmma_f32_16x16x4_f32(
            false, A, false, B, (short)0, acc, false, false);
    }

    // D tile: VGPR r holds row (r + 8*half), cols = lane&15.
    float* outp = out + (size_t)(i0 + 8 * half) * PTOT + p0 + Ncol;
    #pragma unroll
    for (int r = 0; r < 8; ++r) {
        outp[(size_t)r * PTOT] = acc[r];                // immediate offsets
    }
}

extern "C" void kernel_launch(void* const* d_in, const int* in_sizes, int n_in,
                              void* d_out, int out_size, void* d_ws, size_t ws_size,
                              hipStream_t stream) {
    const float* x  = (const float*)d_in[0];   // 150528 floats
    const float* Wt = (const float*)d_in[1];   // 27648 floats
    float* out      = (float*)d_out;           // 602112 floats

    // 12 i-tiles x 49 p-groups; block = 4 waves = 4 p-tiles of one i-tile
    conv_unfold_wmma_f32<<<dim3(12 * 49), dim3(128), 0, stream>>>(x, Wt, out);
}